// CentroidInstanceLoss_90812788507166
// MI455X (gfx1250) — compile-verified
//
#include <hip/hip_runtime.h>
#include <stdint.h>

namespace {
constexpr int   kN      = 2097152;
constexpr int   kD      = 16;
constexpr int   kB      = 8;
constexpr int   kM      = 32;
constexpr int   kG      = kB * kM;            // 256 groups
constexpr float kDeltaV = 0.5f;
constexpr float kDeltaD = 1.5f;
constexpr float kEps    = 1e-8f;

// workspace layout (float offsets)
constexpr int WS_MUSSUM = 0;                   // kG*kD   segment sums
constexpr int WS_CNT    = WS_MUSSUM + kG * kD; // kG      counts
constexpr int WS_MUS    = WS_CNT + kG;         // kG*kD   centroids
constexpr int WS_INV    = WS_MUS + kG * kD;    // kG      1/(M*count)
constexpr int WS_PUSH   = WS_INV + kG;         // 1       L_push (scaled)
constexpr int WS_PART   = WS_PUSH + 64;        // 512     L_pull block partials
constexpr int K1_BLOCKS = 128;
constexpr int K3_BLOCKS = 512;
constexpr int TAB_FLOATS = kG * kD + kG;       // 4352 (mus then inv table)
}

typedef __attribute__((ext_vector_type(2))) float v2f;
typedef __attribute__((ext_vector_type(8))) float v8f;

__device__ __forceinline__ float sq4(float4 a) {
  return a.x * a.x + a.y * a.y + a.z * a.z + a.w * a.w;
}

// ---------------- K0: zero accumulators ----------------
__global__ void cil_k0_zero(float* __restrict__ ws) {
  int i = blockIdx.x * blockDim.x + threadIdx.x;
  if (i < WS_MUS) ws[i] = 0.0f;   // zero segment sums + counts
}

// ---------------- K1: normalized segment sums ----------------
__global__ __launch_bounds__(256) void cil_k1_segsum(
    const float* __restrict__ outp, const int* __restrict__ labels,
    const int* __restrict__ subb, float* __restrict__ ws) {
  __shared__ float lsum[kG * kD];   // 16 KB
  __shared__ float lcnt[kG];
  for (int i = threadIdx.x; i < kG * kD; i += 256) lsum[i] = 0.0f;
  lcnt[threadIdx.x] = 0.0f;         // blockDim == 256 == kG
  __syncthreads();

  const long stride = (long)gridDim.x * blockDim.x;
  for (long p = (long)blockIdx.x * blockDim.x + threadIdx.x; p < kN; p += stride) {
    const float4* g4 = reinterpret_cast<const float4*>(outp + p * kD);
    float4 a = g4[0], b = g4[1], c = g4[2], e = g4[3];
    float n2 = sq4(a) + sq4(b) + sq4(c) + sq4(e);
    float inv = 1.0f / (sqrtf(n2) + kEps);
    int seg = subb[p] * kM + labels[p];
    float* t = &lsum[seg * kD];
    atomicAdd(&t[0],  a.x * inv); atomicAdd(&t[1],  a.y * inv);
    atomicAdd(&t[2],  a.z * inv); atomicAdd(&t[3],  a.w * inv);
    atomicAdd(&t[4],  b.x * inv); atomicAdd(&t[5],  b.y * inv);
    atomicAdd(&t[6],  b.z * inv); atomicAdd(&t[7],  b.w * inv);
    atomicAdd(&t[8],  c.x * inv); atomicAdd(&t[9],  c.y * inv);
    atomicAdd(&t[10], c.z * inv); atomicAdd(&t[11], c.w * inv);
    atomicAdd(&t[12], e.x * inv); atomicAdd(&t[13], e.y * inv);
    atomicAdd(&t[14], e.z * inv); atomicAdd(&t[15], e.w * inv);
    atomicAdd(&lcnt[seg], 1.0f);
  }
  __syncthreads();
  for (int i = threadIdx.x; i < kG * kD; i += 256)
    atomicAdd(&ws[WS_MUSSUM + i], lsum[i]);
  atomicAdd(&ws[WS_CNT + threadIdx.x], lcnt[threadIdx.x]);
}

// ---------------- K2: centroids + L_push (single block) ----------------
__global__ __launch_bounds__(256) void cil_k2_mus_push(float* __restrict__ ws) {
  __shared__ float smu[kG * kD];
  __shared__ float red[256];
  const int g = threadIdx.x;                 // one group per thread
  float cnt = ws[WS_CNT + g];
  float rc  = (cnt > 0.0f) ? (1.0f / cnt) : 0.0f;
#pragma unroll
  for (int d = 0; d < kD; ++d) {
    float m = ws[WS_MUSSUM + g * kD + d] * rc;
    smu[g * kD + d] = m;
    ws[WS_MUS + g * kD + d] = m;
  }
  ws[WS_INV + g] = (cnt > 0.0f) ? (1.0f / ((float)kM * cnt)) : 0.0f;
  __syncthreads();

  const int b = g >> 5, i = g & 31;
  const float* mi = &smu[(b * kM + i) * kD];
  float loc = 0.0f;
  for (int j = 0; j < kM; ++j) {
    if (j == i) continue;
    const float* mj = &smu[(b * kM + j) * kD];
    float pd = 0.0f;
#pragma unroll
    for (int d = 0; d < kD; ++d) pd += fabsf(mi[d] - mj[d]);
    float h = fmaxf(2.0f * kDeltaD - pd, 0.0f);
    loc += h * h;
  }
  red[g] = loc;
  __syncthreads();
  for (int st = 128; st > 0; st >>= 1) {
    if (g < st) red[g] += red[g + st];
    __syncthreads();
  }
  if (g == 0) ws[WS_PUSH] = red[0] * (1.0f / (float)(kM * (kM - 1)));
}

// ---------------- K3: L_pull streaming pass ----------------
__global__ __launch_bounds__(256) void cil_k3_pull(
    const float* __restrict__ outp, const int* __restrict__ labels,
    const int* __restrict__ subb, float* __restrict__ ws) {
  __shared__ float stab[TAB_FLOATS];   // centroids + 1/(M*cnt) table, 17 KB
  __shared__ float red[256];

  // Broadcast table into LDS with gfx1250 async-to-LDS loads (128b per lane).
  const float* tab = ws + WS_MUS;
  for (int idx = threadIdx.x; idx < TAB_FLOATS / 4; idx += 256) {
    unsigned ldsOff = (unsigned)(uintptr_t)(&stab[idx * 4]);  // low 32b = LDS offset
    const float* gp = tab + idx * 4;
    asm volatile("global_load_async_to_lds_b128 %0, %1, off"
                 :: "v"(ldsOff), "v"(gp) : "memory");
  }
  asm volatile("s_wait_asynccnt 0x0" ::: "memory");
  __syncthreads();

  const long stride = (long)gridDim.x * blockDim.x;
  float loc = 0.0f;
  for (long p = (long)blockIdx.x * blockDim.x + threadIdx.x; p < kN; p += stride) {
    const float4* g4 = reinterpret_cast<const float4*>(outp + p * kD);
    float4 a = g4[0], b = g4[1], c = g4[2], e = g4[3];
    float n2 = sq4(a) + sq4(b) + sq4(c) + sq4(e);
    float inv = 1.0f / (sqrtf(n2) + kEps);
    int seg = subb[p] * kM + labels[p];
    const float* mu = &stab[seg * kD];
    float dist =
        fabsf(mu[0]  - a.x * inv) + fabsf(mu[1]  - a.y * inv) +
        fabsf(mu[2]  - a.z * inv) + fabsf(mu[3]  - a.w * inv) +
        fabsf(mu[4]  - b.x * inv) + fabsf(mu[5]  - b.y * inv) +
        fabsf(mu[6]  - b.z * inv) + fabsf(mu[7]  - b.w * inv) +
        fabsf(mu[8]  - c.x * inv) + fabsf(mu[9]  - c.y * inv) +
        fabsf(mu[10] - c.z * inv) + fabsf(mu[11] - c.w * inv) +
        fabsf(mu[12] - e.x * inv) + fabsf(mu[13] - e.y * inv) +
        fabsf(mu[14] - e.z * inv) + fabsf(mu[15] - e.w * inv);
    float h = fmaxf(dist - kDeltaV, 0.0f);
    loc += h * h * stab[kG * kD + seg];       // * 1/(M*count[seg])
  }
  red[threadIdx.x] = loc;
  __syncthreads();
  for (int st = 128; st > 0; st >>= 1) {
    if (threadIdx.x < st) red[threadIdx.x] += red[threadIdx.x + st];
    __syncthreads();
  }
  if (threadIdx.x == 0) ws[WS_PART + blockIdx.x] = red[0];
}

// ---------------- K4: final reduce via WMMA ones-matmul ----------------
__global__ __launch_bounds__(256) void cil_k4_final(const float* __restrict__ ws,
                                                    float* __restrict__ out) {
  __shared__ float s[256];
  const int t = threadIdx.x;
  s[t] = ws[WS_PART + t] + ws[WS_PART + 256 + t];   // 512 partials -> 256
  __syncthreads();
  if (t < 32) {                     // wave-uniform: EXEC all-ones for WMMA
    float x = 0.0f;
#pragma unroll
    for (int k = 0; k < 8; ++k) x += s[t * 8 + k];  // 256 -> 32 (one per lane)
    // C = Ones(16x4) * B(4x16): every row of C holds column sums of B.
    v2f a; a.x = 1.0f; a.y = 1.0f;
    v2f bm; bm.x = x; bm.y = 0.0f;  // 32 live values + 32 zeros across the wave
    v8f c = {};
    c = __builtin_amdgcn_wmma_f32_16x16x4_f32(false, a, false, bm,
                                              (short)0, c, false, false);
    float r = c[0];                 // rows M=0 (lanes 0-15) and M=8 (lanes 16-31)
#pragma unroll
    for (int off = 16; off >= 1; off >>= 1) r += __shfl_xor(r, off, 32);
    if (t == 0) {
      float pull = r * 0.5f;        // both rows counted -> halve
      float push = ws[WS_PUSH];
      out[0] = (pull + push) * (1.0f / (float)kB);
    }
  }
}

extern "C" void kernel_launch(void* const* d_in, const int* in_sizes, int n_in,
                              void* d_out, int out_size, void* d_ws, size_t ws_size,
                              hipStream_t stream) {
  (void)in_sizes; (void)n_in; (void)out_size; (void)ws_size;
  const float* outputs = (const float*)d_in[0];
  const int*   labels  = (const int*)d_in[1];
  const int*   subb    = (const int*)d_in[2];
  float* ws  = (float*)d_ws;
  float* out = (float*)d_out;

  cil_k0_zero<<<(WS_MUS + 255) / 256, 256, 0, stream>>>(ws);
  cil_k1_segsum<<<K1_BLOCKS, 256, 0, stream>>>(outputs, labels, subb, ws);
  cil_k2_mus_push<<<1, 256, 0, stream>>>(ws);
  cil_k3_pull<<<K3_BLOCKS, 256, 0, stream>>>(outputs, labels, subb, ws);
  cil_k4_final<<<1, 256, 0, stream>>>(ws, out);
}